// GraphSAGELayer_39118562132380
// MI455X (gfx1250) — compile-verified
//
#include <hip/hip_runtime.h>

#define N_SUB    50000
#define K_NBR    20
#define IN_DIM   256
#define OUT_DIM  256
#define CAT_DIM  512

#define MTILE    80            // rows per block: 50000 = 625 * 80
#define MSUB     (MTILE / 16)  // 5 sub-tiles of 16 rows

typedef float v2f __attribute__((ext_vector_type(2)));
typedef float v8f __attribute__((ext_vector_type(8)));

// ---------------------------------------------------------------------------
// Kernel 1: neighbor gather + mean.  One block per output row, 64 lanes x f4.
// X_sub (51 MB) is L2-resident (192 MB L2): ~1.0 GB of L2-side reads total.
// ---------------------------------------------------------------------------
__global__ __launch_bounds__(64) void gather_mean_kernel(
    const float* __restrict__ X, const int* __restrict__ sub_nodes,
    const int* __restrict__ nbr_ids, const int* __restrict__ g2s,
    float* __restrict__ M)
{
    __shared__ int sidx[K_NBR];
    const int row = blockIdx.x;
    const int tid = threadIdx.x;
    if (tid < K_NBR) {
        const int node = sub_nodes[row];
        const int g    = nbr_ids[node * K_NBR + tid];
        const int s    = g2s[g];
        sidx[tid] = (s < 0) ? row : s;
    }
    __syncthreads();
    const float4* Xv = (const float4*)X;
    float4 acc = make_float4(0.f, 0.f, 0.f, 0.f);
#pragma unroll
    for (int k = 0; k < K_NBR; ++k) {
        const float4 v = Xv[(size_t)sidx[k] * (IN_DIM / 4) + tid];
        acc.x += v.x; acc.y += v.y; acc.z += v.z; acc.w += v.w;
    }
    const float s = 1.0f / (float)K_NBR;
    float4 o;
    o.x = acc.x * s; o.y = acc.y * s; o.z = acc.z * s; o.w = acc.w * s;
    ((float4*)M)[(size_t)row * (IN_DIM / 4) + tid] = o;
}

// ---------------------------------------------------------------------------
// Kernel 2: H = relu(A @ W^T + b)   (M=N_SUB, K=256, N=256), fp32 WMMA.
// Block = 256 threads (8 waves) owns an 80-row M tile (5 sub-tiles of 16).
// Per 64-wide K chunk, each wave caches its B fragments in registers once
// (v2f B0[16], B1[16]) and reuses them across all 5 A sub-tiles from LDS:
// 160 wmma per chunk per wave per 8 KB of B  -> 5x less weight traffic than
// a 16-row tile.  Safe in-place (C == A): block only touches its own rows.
// ---------------------------------------------------------------------------
__global__ __launch_bounds__(256) void gemm_bias_relu_kernel(
    const float* __restrict__ A, const float* __restrict__ W,
    const float* __restrict__ bias, float* __restrict__ C, int Ka)
{
    __shared__ float lA[MTILE][68];         // +4 pad: conflict-free wmma reads
    const int m0    = blockIdx.x * MTILE;
    const int tid   = threadIdx.x;
    const int wave  = tid >> 5;
    const int lane  = tid & 31;
    const int laneM = lane & 15;
    const int khalf = (lane >> 4) * 2;      // 0 for lanes 0-15, 2 for 16-31
    const int n0    = wave * 32;

    v8f acc0[MSUB], acc1[MSUB];
#pragma unroll
    for (int mt = 0; mt < MSUB; ++mt) { acc0[mt] = (v8f){}; acc1[mt] = (v8f){}; }

#pragma unroll 1
    for (int kc = 0; kc < Ka; kc += 64) {
        // cooperative 80x64 A-tile load: 5 float4 per thread
#pragma unroll
        for (int i = 0; i < MSUB; ++i) {
            const int r  = i * 16 + (tid >> 4);
            const int c4 = (tid & 15) * 4;
            const float4 v = *(const float4*)(A + (size_t)(m0 + r) * Ka + kc + c4);
            lA[r][c4 + 0] = v.x; lA[r][c4 + 1] = v.y;
            lA[r][c4 + 2] = v.z; lA[r][c4 + 3] = v.w;
        }
        __syncthreads();

        // stage this wave's B fragments for the whole K chunk in registers
        v2f B0[16], B1[16];
#pragma unroll
        for (int ks = 0; ks < 16; ++ks) {
            const float* wp0 = W + (size_t)(n0 + laneM) * Ka + kc + 4 * ks + khalf;
            const float* wp1 = wp0 + (size_t)16 * Ka;
            B0[ks].x = wp0[0]; B0[ks].y = wp0[1];
            B1[ks].x = wp1[0]; B1[ks].y = wp1[1];
        }

#pragma unroll
        for (int mt = 0; mt < MSUB; ++mt) {
#pragma unroll
            for (int ks = 0; ks < 16; ++ks) {
                v2f a;
                a.x = lA[mt * 16 + laneM][4 * ks + khalf];
                a.y = lA[mt * 16 + laneM][4 * ks + khalf + 1];
                acc0[mt] = __builtin_amdgcn_wmma_f32_16x16x4_f32(
                    false, a, false, B0[ks], (short)0, acc0[mt], false, false);
                acc1[mt] = __builtin_amdgcn_wmma_f32_16x16x4_f32(
                    false, a, false, B1[ks], (short)0, acc1[mt], false, false);
            }
        }
        __syncthreads();
    }

    // epilogue: bias + relu + store
    const int   hi  = (lane >> 4) * 8;
    const float bb0 = bias[n0 + laneM];
    const float bb1 = bias[n0 + 16 + laneM];
#pragma unroll
    for (int mt = 0; mt < MSUB; ++mt) {
#pragma unroll
        for (int r = 0; r < 8; ++r) {
            float v0 = acc0[mt][r] + bb0; v0 = v0 > 0.f ? v0 : 0.f;
            float v1 = acc1[mt][r] + bb1; v1 = v1 > 0.f ? v1 : 0.f;
            const size_t row = (size_t)(m0 + mt * 16 + r + hi);
            C[row * OUT_DIM + n0 + laneM]      = v0;
            C[row * OUT_DIM + n0 + 16 + laneM] = v1;
        }
    }
}

// ---------------------------------------------------------------------------
// Kernel 3: out = l2norm(relu(X @ Wx^T + H @ Wh^T + b_final))
// Wx = W_final[:, 0:256], Wh = W_final[:, 256:512] (row stride CAT_DIM).
// Same 80-row blocking; block owns complete output rows, so bias + relu +
// row L2-normalization fuse in-block via LDS f32 atomics.
// ---------------------------------------------------------------------------
__global__ __launch_bounds__(256) void final_kernel(
    const float* __restrict__ X, const float* __restrict__ H,
    const float* __restrict__ Wf, const float* __restrict__ bf,
    float* __restrict__ out)
{
    __shared__ float lA[MTILE][68];
    __shared__ float rowsq[MTILE];
    const int m0    = blockIdx.x * MTILE;
    const int tid   = threadIdx.x;
    const int wave  = tid >> 5;
    const int lane  = tid & 31;
    const int laneM = lane & 15;
    const int khalf = (lane >> 4) * 2;
    const int n0    = wave * 32;

    v8f acc0[MSUB], acc1[MSUB];
#pragma unroll
    for (int mt = 0; mt < MSUB; ++mt) { acc0[mt] = (v8f){}; acc1[mt] = (v8f){}; }

#pragma unroll 1
    for (int phase = 0; phase < 2; ++phase) {
        const float* A     = phase ? H : X;
        const int    wkoff = phase * IN_DIM;   // column offset inside W_final
#pragma unroll 1
        for (int kc = 0; kc < IN_DIM; kc += 64) {
#pragma unroll
            for (int i = 0; i < MSUB; ++i) {
                const int r  = i * 16 + (tid >> 4);
                const int c4 = (tid & 15) * 4;
                const float4 v = *(const float4*)(A + (size_t)(m0 + r) * IN_DIM + kc + c4);
                lA[r][c4 + 0] = v.x; lA[r][c4 + 1] = v.y;
                lA[r][c4 + 2] = v.z; lA[r][c4 + 3] = v.w;
            }
            __syncthreads();

            v2f B0[16], B1[16];
#pragma unroll
            for (int ks = 0; ks < 16; ++ks) {
                const float* wp0 =
                    Wf + (size_t)(n0 + laneM) * CAT_DIM + wkoff + kc + 4 * ks + khalf;
                const float* wp1 = wp0 + (size_t)16 * CAT_DIM;
                B0[ks].x = wp0[0]; B0[ks].y = wp0[1];
                B1[ks].x = wp1[0]; B1[ks].y = wp1[1];
            }

#pragma unroll
            for (int mt = 0; mt < MSUB; ++mt) {
#pragma unroll
                for (int ks = 0; ks < 16; ++ks) {
                    v2f a;
                    a.x = lA[mt * 16 + laneM][4 * ks + khalf];
                    a.y = lA[mt * 16 + laneM][4 * ks + khalf + 1];
                    acc0[mt] = __builtin_amdgcn_wmma_f32_16x16x4_f32(
                        false, a, false, B0[ks], (short)0, acc0[mt], false, false);
                    acc1[mt] = __builtin_amdgcn_wmma_f32_16x16x4_f32(
                        false, a, false, B1[ks], (short)0, acc1[mt], false, false);
                }
            }
            __syncthreads();
        }
    }

    // bias + relu, accumulate per-row sum of squares in LDS
    if (tid < MTILE) rowsq[tid] = 0.f;
    __syncthreads();
    const int   hi  = (lane >> 4) * 8;
    const float bb0 = bf[n0 + laneM];
    const float bb1 = bf[n0 + 16 + laneM];
    float v0s[MSUB][8], v1s[MSUB][8];
#pragma unroll
    for (int mt = 0; mt < MSUB; ++mt) {
#pragma unroll
        for (int r = 0; r < 8; ++r) {
            float v0 = acc0[mt][r] + bb0; v0 = v0 > 0.f ? v0 : 0.f;
            float v1 = acc1[mt][r] + bb1; v1 = v1 > 0.f ? v1 : 0.f;
            v0s[mt][r] = v0; v1s[mt][r] = v1;
            atomicAdd(&rowsq[mt * 16 + r + hi], v0 * v0 + v1 * v1);  // ds_add_f32
        }
    }
    __syncthreads();
#pragma unroll
    for (int mt = 0; mt < MSUB; ++mt) {
#pragma unroll
        for (int r = 0; r < 8; ++r) {
            const int   rl    = mt * 16 + r + hi;
            const float denom = fmaxf(sqrtf(rowsq[rl]), 1e-12f);
            const float inv   = 1.0f / denom;
            const size_t row  = (size_t)(m0 + rl);
            out[row * OUT_DIM + n0 + laneM]      = v0s[mt][r] * inv;
            out[row * OUT_DIM + n0 + 16 + laneM] = v1s[mt][r] * inv;
        }
    }
}

// ---------------------------------------------------------------------------
extern "C" void kernel_launch(void* const* d_in, const int* in_sizes, int n_in,
                              void* d_out, int out_size, void* d_ws, size_t ws_size,
                              hipStream_t stream) {
    const float* X_sub     = (const float*)d_in[0];
    const int*   sub_nodes = (const int*)  d_in[1];
    const int*   nbr_ids   = (const int*)  d_in[2];
    const int*   g2s       = (const int*)  d_in[3];
    const float* W_nbr     = (const float*)d_in[4];
    const float* b_nbr     = (const float*)d_in[5];
    const float* W_final   = (const float*)d_in[6];
    const float* b_final   = (const float*)d_in[7];
    float*       out       = (float*)d_out;

    // Single 50000x256 f32 scratch buffer: holds nbr_mean, then (in-place) h_nbr.
    float* Mbuf = (float*)d_ws;

    gather_mean_kernel<<<N_SUB, 64, 0, stream>>>(X_sub, sub_nodes, nbr_ids, g2s, Mbuf);
    gemm_bias_relu_kernel<<<N_SUB / MTILE, 256, 0, stream>>>(Mbuf, W_nbr, b_nbr, Mbuf, IN_DIM);
    final_kernel<<<N_SUB / MTILE, 256, 0, stream>>>(X_sub, Mbuf, W_final, b_final, out);
}